// L1_78769700208928
// MI455X (gfx1250) — compile-verified
//
#include <hip/hip_runtime.h>
#include <stdint.h>

// Negative pairwise L1 distance on MI455X (gfx1250, wave32).
// out[n,m] = -sum_d |x[n,d]-y[m,d]|, N=M=2048, D=128, fp32.
//
// VALU-bound (~57 ops/byte); WMMA inapplicable (|a-b| not bilinear).
// CDNA5 path: software-pipelined async global->LDS staging (ASYNCcnt,
// s_wait_asynccnt N with double-buffered dynamic LDS), conflict-free
// padded LDS, 8x8 register tile per thread, VOPD-friendly inner loop.

#define TILE    128   // output tile is TILE x TILE per block
#define DC      32    // D-chunk staged in LDS
#define LDS_STR 36    // DC + 4 pad floats: 16B-aligned rows, conflict-free
#define THREADS 256   // 8 waves (wave32)
#define NCHUNK  4     // D / DC
#define BUF_FLOATS (2 * TILE * LDS_STR)          // xs+ys per buffer
#define LDS_BYTES  (2 * BUF_FLOATS * 4)          // double buffered: 73728 B

#if __has_builtin(__builtin_amdgcn_global_load_async_to_lds_b128)
#define HAVE_ASYNC_B128 1
#else
#define HAVE_ASYNC_B128 0
#endif

typedef int v4i __attribute__((__vector_size__(4 * sizeof(int))));
typedef __attribute__((address_space(1))) v4i* global_v4i_p;
typedef __attribute__((address_space(3))) v4i* lds_v4i_p;

__device__ __forceinline__ void g2l_b128(const float* g, float* l) {
#if HAVE_ASYNC_B128
    // Per-lane async copy of 16B global -> LDS, tracked by ASYNCcnt.
    __builtin_amdgcn_global_load_async_to_lds_b128(
        (global_v4i_p)(uintptr_t)g,
        (lds_v4i_p)(uintptr_t)l,
        /*offset=*/0, /*cpol=*/0);
#else
    *(float4*)l = *(const float4*)g;
#endif
}

template <int N>
__device__ __forceinline__ void wait_asynccnt_le() {
#if HAVE_ASYNC_B128
#if __has_builtin(__builtin_amdgcn_s_wait_asynccnt)
    __builtin_amdgcn_s_wait_asynccnt(N);
#else
    asm volatile("s_wait_asynccnt %0" :: "i"(N) : "memory");
#endif
#endif
}

// Stage one D-chunk (128 rows x 32 dims of x and y) into LDS buffer `buf`.
// 8 async b128 instructions per thread (ASYNCcnt += 8 per wave).
__device__ __forceinline__ void stage_chunk(const float* __restrict__ x,
                                            const float* __restrict__ y,
                                            float* __restrict__ xs,
                                            float* __restrict__ ys,
                                            int rowBase, int colBase,
                                            int dBase, int tid) {
    const int D = 128;
#pragma unroll
    for (int k = 0; k < 4; ++k) {
        int idx = tid + THREADS * k;      // 0..1023
        int r   = idx >> 3;               // row 0..127
        int c   = (idx & 7) << 2;         // float col 0..28 step 4
        g2l_b128(x + (size_t)(rowBase + r) * D + dBase + c, &xs[r * LDS_STR + c]);
        g2l_b128(y + (size_t)(colBase + r) * D + dBase + c, &ys[r * LDS_STR + c]);
    }
}

__global__ __launch_bounds__(THREADS)
void l1neg_kernel(const float* __restrict__ x,
                  const float* __restrict__ y,
                  float* __restrict__ out,
                  int Mtot)
{
    extern __shared__ float smem[];      // 2 buffers x (xs + ys)

    const int tid     = threadIdx.x;
    const int rowBase = blockIdx.y * TILE;
    const int colBase = blockIdx.x * TILE;

    const int cx = tid & 15;   // output col lane within tile
    const int cy = tid >> 4;   // output row lane within tile

    float acc[8][8];
#pragma unroll
    for (int a = 0; a < 8; ++a)
#pragma unroll
        for (int b = 0; b < 8; ++b) acc[a][b] = 0.0f;

    // Prologue: kick off chunk 0 into buffer 0.
    stage_chunk(x, y, smem, smem + TILE * LDS_STR, rowBase, colBase, 0, tid);

    for (int c = 0; c < NCHUNK; ++c) {
        float* xs = smem + (c & 1) * BUF_FLOATS;
        float* ys = xs + TILE * LDS_STR;

        if (c + 1 < NCHUNK) {
            // Prefetch next chunk into the other buffer BEFORE waiting.
            // Its last readers (chunk c-1 compute) passed the barrier below.
            float* xs2 = smem + ((c + 1) & 1) * BUF_FLOATS;
            stage_chunk(x, y, xs2, xs2 + TILE * LDS_STR,
                        rowBase, colBase, (c + 1) * DC, tid);
            // Async ops complete in order: <=8 outstanding => chunk c done,
            // chunk c+1 still in flight behind the compute below.
            wait_asynccnt_le<8>();
        } else {
            wait_asynccnt_le<0>();
        }
        __syncthreads();   // chunk c visible from all waves

        // Compute: per d-quad, 16 ds_load_b128 feed 512 VALU ops.
#pragma unroll
        for (int d = 0; d < DC; d += 4) {
            float4 xr[8], yr[8];
#pragma unroll
            for (int a = 0; a < 8; ++a)
                xr[a] = *(const float4*)&xs[(cy + 16 * a) * LDS_STR + d];
#pragma unroll
            for (int b = 0; b < 8; ++b)
                yr[b] = *(const float4*)&ys[(cx + 16 * b) * LDS_STR + d];
#pragma unroll
            for (int a = 0; a < 8; ++a)
#pragma unroll
                for (int b = 0; b < 8; ++b) {
                    // v_sub_f32 + v_add_f32 |src| (VOPD dual-issue friendly)
                    acc[a][b] += fabsf(xr[a].x - yr[b].x);
                    acc[a][b] += fabsf(xr[a].y - yr[b].y);
                    acc[a][b] += fabsf(xr[a].z - yr[b].z);
                    acc[a][b] += fabsf(xr[a].w - yr[b].w);
                }
        }

        if (c + 2 < NCHUNK)
            __syncthreads();  // buf (c&1) must be fully read before chunk c+2 issue
    }

    // Write 8x8 outputs; lanes with consecutive cx give coalesced stores.
#pragma unroll
    for (int a = 0; a < 8; ++a) {
        const size_t row = (size_t)(rowBase + cy + 16 * a);
#pragma unroll
        for (int b = 0; b < 8; ++b) {
            const int col = colBase + cx + 16 * b;
            out[row * (size_t)Mtot + col] = -acc[a][b];
        }
    }
}

extern "C" void kernel_launch(void* const* d_in, const int* in_sizes, int n_in,
                              void* d_out, int out_size, void* d_ws, size_t ws_size,
                              hipStream_t stream) {
    const float* x = (const float*)d_in[0];
    const float* y = (const float*)d_in[1];
    float* out = (float*)d_out;

    const int D = 128;
    const int N = in_sizes[0] / D;   // 2048
    const int M = in_sizes[1] / D;   // 2048

    dim3 grid(M / TILE, N / TILE);   // 16 x 16 blocks
    l1neg_kernel<<<grid, THREADS, LDS_BYTES, stream>>>(x, y, out, M);
}